// SelectiveSSM_24893630447892
// MI455X (gfx1250) — compile-verified
//
#include <hip/hip_runtime.h>
#include <math.h>

typedef __attribute__((ext_vector_type(2))) float v2f;
typedef __attribute__((ext_vector_type(8))) float v8f;

#define DM   1024   // d_model
#define DS   16     // d_state
#define DTR  64     // dt_rank
#define NPJ  96     // dt_rank + 2*d_state
#define BSZ  2
#define TLEN 2048
#define MROWS (BSZ*TLEN)   // 4096
#define NC   32            // time chunks
#define CL   (TLEN/NC)     // 64 steps per chunk

// ---------------------------------------------------------------------------
// GEMM 1: x_proj[4096 x 96] = x[4096 x 1024] @ W_x[1024 x 96]
// One wave per (16-row slab x 48 cols): 3 N-tiles register-blocked so each
// A float2 feeds 3 WMMAs (x read once per 48 output cols instead of per 16).
// V_WMMA_F32_16X16X4_F32; f32 A layout: lane<16 -> {k0,k0+1} of row=lane,
// lane>=16 -> {k0+2,k0+3}; B mirrored; C/D in documented 8-VGPR layout.
// ---------------------------------------------------------------------------
__global__ void proj_gemm_kernel(const float* __restrict__ x,
                                 const float* __restrict__ Wx,
                                 float* __restrict__ xproj) {
    const int m0   = blockIdx.x * 16;
    const int nb   = blockIdx.y * 48;        // covers n-tiles nb, nb+16, nb+32
    const int lane = threadIdx.x;
    const int half = lane >> 4;              // 0: lanes 0-15, 1: lanes 16-31
    const int l16  = lane & 15;

    v8f acc0 = {}, acc1 = {}, acc2 = {};
    const float* arow = x + (size_t)(m0 + l16) * DM;

    for (int k0 = 0; k0 < DM; k0 += 4) {
        const int ka = k0 + 2 * half;
        v2f a = *(const v2f*)(arow + ka);
        const float* bp = Wx + (size_t)ka * NPJ + nb + l16;
        v2f b0, b1, b2;
        b0.x = bp[0];        b0.y = bp[NPJ];
        b1.x = bp[16];       b1.y = bp[16 + NPJ];
        b2.x = bp[32];       b2.y = bp[32 + NPJ];
        acc0 = __builtin_amdgcn_wmma_f32_16x16x4_f32(false, a, false, b0, (short)0, acc0, false, false);
        acc1 = __builtin_amdgcn_wmma_f32_16x16x4_f32(false, a, false, b1, (short)0, acc1, false, false);
        acc2 = __builtin_amdgcn_wmma_f32_16x16x4_f32(false, a, false, b2, (short)0, acc2, false, false);
    }
#pragma unroll
    for (int v = 0; v < 8; ++v) {
        const int row = m0 + v + (half ? 8 : 0);
        float* orow = xproj + (size_t)row * NPJ + nb + l16;
        orow[0]  = acc0[v];
        orow[16] = acc1[v];
        orow[32] = acc2[v];
    }
}

// ---------------------------------------------------------------------------
// GEMM 2 + softplus: dt[4096 x 1024] = softplus(xproj[:, :64] @ W_dt[64 x 1024] + b_dt)
// 4 N-tiles register-blocked per wave; K=64 -> 16 WMMA steps fully unrolled.
// ---------------------------------------------------------------------------
__global__ void dt_gemm_kernel(const float* __restrict__ xproj,
                               const float* __restrict__ Wdt,
                               const float* __restrict__ bdt,
                               float* __restrict__ dt) {
    const int m0   = blockIdx.x * 16;
    const int nb   = blockIdx.y * 64;        // covers n-tiles nb .. nb+48
    const int lane = threadIdx.x;
    const int half = lane >> 4;
    const int l16  = lane & 15;

    v8f acc[4] = {};
    const float* arow = xproj + (size_t)(m0 + l16) * NPJ;   // dt_raw = first 64 cols

#pragma unroll
    for (int k0 = 0; k0 < DTR; k0 += 4) {
        const int ka = k0 + 2 * half;
        v2f a = *(const v2f*)(arow + ka);
        const float* bp = Wdt + (size_t)ka * DM + nb + l16;
#pragma unroll
        for (int j = 0; j < 4; ++j) {
            v2f b;
            b.x = bp[j * 16];
            b.y = bp[j * 16 + DM];
            acc[j] = __builtin_amdgcn_wmma_f32_16x16x4_f32(false, a, false, b, (short)0, acc[j], false, false);
        }
    }
#pragma unroll
    for (int v = 0; v < 8; ++v) {
        const int row = m0 + v + (half ? 8 : 0);
#pragma unroll
        for (int j = 0; j < 4; ++j) {
            const int col = nb + j * 16 + l16;
            float z = acc[j][v] + bdt[col];
            float sp = (z > 20.0f) ? z : log1pf(__expf(z));   // stable softplus
            dt[(size_t)row * DM + col] = sp;
        }
    }
}

// ---------------------------------------------------------------------------
// Scan pass 1: per (b, chunk, d) compute P = prod(A_bar) and S = local scan end
// state (zero entry state) per state s. One lane per channel d; 64 steps.
// ---------------------------------------------------------------------------
__global__ void scan_chunk1_kernel(const float* __restrict__ x,
                                   const float* __restrict__ xproj,
                                   const float* __restrict__ dt,
                                   const float* __restrict__ A_log,
                                   float* __restrict__ P,
                                   float* __restrict__ S) {
    const int bc = blockIdx.x;            // b*NC + c
    const int b  = bc / NC;
    const int c  = bc % NC;
    const int d  = blockIdx.y * blockDim.x + threadIdx.x;   // 0..DM-1

    float An[DS];
#pragma unroll
    for (int s = 0; s < DS; ++s) An[s] = -__expf(A_log[d * DS + s]);

    float p[DS], h[DS];
#pragma unroll
    for (int s = 0; s < DS; ++s) { p[s] = 1.0f; h[s] = 0.0f; }

    const int t0 = c * CL;
    for (int t = t0; t < t0 + CL; ++t) {
        const int bt = b * TLEN + t;
        const float dtv = dt[(size_t)bt * DM + d];
        const float xv  = x[(size_t)bt * DM + d];
        const float dtx = dtv * xv;
        const float* Bm = xproj + (size_t)bt * NPJ + DTR;   // block-uniform address
#pragma unroll
        for (int s = 0; s < DS; ++s) {
            const float a = __expf(dtv * An[s]);
            p[s] *= a;
            h[s] = a * h[s] + dtx * Bm[s];
        }
    }
    const size_t base = (((size_t)b * NC + c) * DM + d) * DS;
#pragma unroll
    for (int s = 0; s < DS; ++s) { P[base + s] = p[s]; S[base + s] = h[s]; }
}

// ---------------------------------------------------------------------------
// Carry pass: sequential over the 32 chunks, parallel over B*DM*DS = 32768.
// H0[b,c,d,s] = entry state of chunk c.
// ---------------------------------------------------------------------------
__global__ void scan_carry_kernel(const float* __restrict__ P,
                                  const float* __restrict__ S,
                                  float* __restrict__ H0) {
    const int idx = blockIdx.x * blockDim.x + threadIdx.x;  // b*DM*DS + d*DS + s
    const int b   = idx / (DM * DS);
    const int ds  = idx % (DM * DS);
    float h = 0.0f;
    for (int c = 0; c < NC; ++c) {
        const size_t off = (((size_t)b * NC + c) * DM * DS) + ds;
        H0[off] = h;
        h = P[off] * h + S[off];
    }
}

// ---------------------------------------------------------------------------
// Scan pass 2: replay each chunk from its entry state, emit y = C·h + x*D.
// ---------------------------------------------------------------------------
__global__ void scan_chunk2_kernel(const float* __restrict__ x,
                                   const float* __restrict__ xproj,
                                   const float* __restrict__ dt,
                                   const float* __restrict__ A_log,
                                   const float* __restrict__ Dvec,
                                   const float* __restrict__ H0,
                                   float* __restrict__ out) {
    const int bc = blockIdx.x;
    const int b  = bc / NC;
    const int c  = bc % NC;
    const int d  = blockIdx.y * blockDim.x + threadIdx.x;

    float An[DS];
#pragma unroll
    for (int s = 0; s < DS; ++s) An[s] = -__expf(A_log[d * DS + s]);

    float h[DS];
    const size_t hbase = (((size_t)b * NC + c) * DM + d) * DS;
#pragma unroll
    for (int s = 0; s < DS; ++s) h[s] = H0[hbase + s];

    const float dval = Dvec[d];
    const int t0 = c * CL;
    for (int t = t0; t < t0 + CL; ++t) {
        const int bt = b * TLEN + t;
        const float dtv = dt[(size_t)bt * DM + d];
        const float xv  = x[(size_t)bt * DM + d];
        const float dtx = dtv * xv;
        const float* Bm = xproj + (size_t)bt * NPJ + DTR;
        const float* Cm = xproj + (size_t)bt * NPJ + DTR + DS;
        float y = 0.0f;
#pragma unroll
        for (int s = 0; s < DS; ++s) {
            const float a = __expf(dtv * An[s]);
            h[s] = a * h[s] + dtx * Bm[s];
            y += h[s] * Cm[s];
        }
        out[(size_t)bt * DM + d] = y + xv * dval;
    }
}

// ---------------------------------------------------------------------------
extern "C" void kernel_launch(void* const* d_in, const int* in_sizes, int n_in,
                              void* d_out, int out_size, void* d_ws, size_t ws_size,
                              hipStream_t stream) {
    const float* x     = (const float*)d_in[0];
    const float* Wx    = (const float*)d_in[1];
    const float* Wdt   = (const float*)d_in[2];
    const float* bdt   = (const float*)d_in[3];
    const float* A_log = (const float*)d_in[4];
    const float* Dvec  = (const float*)d_in[5];
    float* out = (float*)d_out;

    // workspace layout (~31 MB total)
    char* ws = (char*)d_ws;
    float* xproj = (float*)ws;                                  //  4096*96   = 1.57 MB
    ws += (size_t)MROWS * NPJ * sizeof(float);
    float* dt    = (float*)ws;                                  //  4096*1024 = 16.8 MB
    ws += (size_t)MROWS * DM * sizeof(float);
    float* P     = (float*)ws;                                  //  2*32*1024*16 = 4.2 MB
    ws += (size_t)BSZ * NC * DM * DS * sizeof(float);
    float* S     = (float*)ws;
    ws += (size_t)BSZ * NC * DM * DS * sizeof(float);
    float* H0    = (float*)ws;

    // GEMM 1: 256 m-tiles x 2 n-groups (3 tiles each), one wave per block
    proj_gemm_kernel<<<dim3(MROWS / 16, 2), 32, 0, stream>>>(x, Wx, xproj);
    // GEMM 2: 256 m-tiles x 16 n-groups (4 tiles each)
    dt_gemm_kernel<<<dim3(MROWS / 16, 16), 32, 0, stream>>>(xproj, Wdt, bdt, dt);
    // Chunked scan: pass 1
    scan_chunk1_kernel<<<dim3(BSZ * NC, DM / 256), 256, 0, stream>>>(x, xproj, dt, A_log, P, S);
    // Carry combine (32768 independent (b,d,s) chains of 32 steps)
    scan_carry_kernel<<<(BSZ * DM * DS) / 256, 256, 0, stream>>>(P, S, H0);
    // Pass 2: replay with correct entry states, emit output
    scan_chunk2_kernel<<<dim3(BSZ * NC, DM / 256), 256, 0, stream>>>(x, xproj, dt, A_log, Dvec, H0, out);
}